// Unet_40k_47845935677650
// MI455X (gfx1250) — compile-verified
//
#include <hip/hip_runtime.h>

typedef __attribute__((ext_vector_type(2))) float v2f;
typedef __attribute__((ext_vector_type(8))) float v8f;

#define TILE_M 64
#define TILE_N 32
#define TILE_K 32

// ---------------------------------------------------------------------------
// Weight transpose + zero-pad:  W (N rows x K cols, row-major)  ->  Wt[Kp][N]
// ---------------------------------------------------------------------------
__global__ void k_wt(const float* __restrict__ W, float* __restrict__ Wt,
                     int K, int Kp, int N) {
  int idx = blockIdx.x * blockDim.x + threadIdx.x;
  if (idx >= Kp * N) return;
  int k = idx / N, n = idx - k * N;
  Wt[idx] = (k < K) ? W[(long)n * K + k] : 0.0f;
}

// ---------------------------------------------------------------------------
// Gather-fused WMMA GEMM:  Y[m, colOff+n] = bias[n] + sum_k A[m,k] * Wt[k][n]
//   A[m,k] = no ? X[no[7m + k/ic]*ic + (k%ic)]  (7-neighbor gather, k < K)
//              : X[m*ic + k]                     (plain GEMM, ic == K)
// Block: 256 threads = 8 waves, tile 64x32, K chunked by 32 through LDS.
// Each wave computes one 16x16 tile with V_WMMA_F32_16X16X4_F32.
// ---------------------------------------------------------------------------
__global__ __launch_bounds__(256) void k_gemm(
    const float* __restrict__ X, const int* __restrict__ no,
    const float* __restrict__ Wt, const float* __restrict__ bias,
    float* __restrict__ Y,
    int M, int K, int Kp, int N, int ic, int ldY, int colOff)
{
  __shared__ __align__(16) float As[TILE_M * 34];  // 64 rows x 32 k (pad 34)
  __shared__ __align__(16) float Bs[TILE_K * 33];  // 32 k x 32 n  (pad 33)

  const int tid  = threadIdx.x;
  const int lane = tid & 31;
  const int wave = tid >> 5;
  const int wr   = wave >> 1;          // 0..3  (row sub-tile)
  const int wc   = wave & 1;           // 0..1  (col sub-tile)
  const int i0   = blockIdx.x * TILE_M;
  const int n0   = blockIdx.y * TILE_N;
  const int nloc = wc * 16 + (lane & 15);          // 0..31 within block tile
  const int koff = (lane >= 16) ? 2 : 0;           // K-pair per lane half
  const int am   = wr * 16 + (lane & 15);          // A row within block tile

  float bv = bias ? bias[n0 + nloc] : 0.0f;
  v8f acc = {bv, bv, bv, bv, bv, bv, bv, bv};

  for (int kc = 0; kc < Kp; kc += TILE_K) {
    __syncthreads();
    // stage A (with fused gather), zero-pad K remainder and M edge
    for (int idx = tid; idx < TILE_M * TILE_K; idx += 256) {
      int m   = idx >> 5;
      int kk  = idx & 31;
      int k   = kc + kk;
      int row = i0 + m;
      float v = 0.0f;
      if (row < M && k < K) {
        if (no) {
          int q = k / ic;
          int r = k - q * ic;
          v = X[(long)no[row * 7 + q] * ic + r];
        } else {
          v = X[(long)row * ic + k];
        }
      }
      As[m * 34 + kk] = v;
    }
    // stage B from pre-transposed, pre-padded Wt
    for (int idx = tid; idx < TILE_K * TILE_N; idx += 256) {
      int kk = idx >> 5;
      int n  = idx & 31;
      Bs[kk * 33 + n] = Wt[(long)(kc + kk) * N + n0 + n];
    }
    __syncthreads();
#pragma unroll
    for (int k2 = 0; k2 < TILE_K; k2 += 4) {
      // A 16x4: lanes 0-15 -> K = k2+0,k2+1 ; lanes 16-31 -> K = k2+2,k2+3
      v2f a = *(const v2f*)(As + am * 34 + k2 + koff);
      // B 4x16: lane%16 = N, lane half selects K pair
      v2f b;
      b.x = Bs[(k2 + koff    ) * 33 + nloc];
      b.y = Bs[(k2 + koff + 1) * 33 + nloc];
      acc = __builtin_amdgcn_wmma_f32_16x16x4_f32(
          false, a, false, b, (short)0, acc, false, false);
    }
  }

  // D layout: VGPR v -> M = v (lanes 0-15) / v+8 (lanes 16-31), N = lane%16
  const int mg0 = i0 + wr * 16 + ((lane >= 16) ? 8 : 0);
  const int col = colOff + n0 + nloc;
#pragma unroll
  for (int v = 0; v < 8; ++v) {
    int mg = mg0 + v;
    if (mg < M) Y[(long)mg * ldY + col] = acc[v];
  }
}

// ---------------------------------------------------------------------------
// BatchNorm stats: one block per channel; deterministic tree reduction.
// ---------------------------------------------------------------------------
__global__ __launch_bounds__(256) void k_bnstats(
    const float* __restrict__ Y, float* __restrict__ mean,
    float* __restrict__ var, int M, int C)
{
  __shared__ float ss[256];
  __shared__ float sq[256];
  int c = blockIdx.x;
  float s = 0.0f, q = 0.0f;
  for (int r = threadIdx.x; r < M; r += 256) {
    float v = Y[(long)r * C + c];
    s += v; q += v * v;
  }
  ss[threadIdx.x] = s; sq[threadIdx.x] = q;
  __syncthreads();
  for (int off = 128; off > 0; off >>= 1) {
    if (threadIdx.x < off) {
      ss[threadIdx.x] += ss[threadIdx.x + off];
      sq[threadIdx.x] += sq[threadIdx.x + off];
    }
    __syncthreads();
  }
  if (threadIdx.x == 0) {
    float mu = ss[0] / (float)M;
    float v  = sq[0] / (float)M - mu * mu;
    mean[c] = mu;
    var[c]  = v > 0.0f ? v : 0.0f;
  }
}

__global__ void k_bnapply(float* __restrict__ Y, const float* __restrict__ mean,
                          const float* __restrict__ var, const float* __restrict__ g,
                          const float* __restrict__ e, int M, int C)
{
  int idx = blockIdx.x * blockDim.x + threadIdx.x;
  if (idx >= M * C) return;
  int c = idx % C;
  float z = (Y[idx] - mean[c]) * rsqrtf(var[c] + 1e-5f) * g[c] + e[c];
  Y[idx] = z >= 0.0f ? z : 0.2f * z;
}

// ---------------------------------------------------------------------------
// Faithful raw-reshape pool: out[i,c] = mean_k flat[i*7*F + c*7 + k],
//   flat[m] = X[no[m/F] * F + (m % F)],  F = 1<<shift
// ---------------------------------------------------------------------------
__global__ void k_pool(const float* __restrict__ X, const int* __restrict__ no,
                       float* __restrict__ out, int nOut, int shift)
{
  int F = 1 << shift;
  int idx = blockIdx.x * blockDim.x + threadIdx.x;
  if (idx >= nOut * F) return;
  int i = idx >> shift;
  int c = idx & (F - 1);
  int m = i * 7 * F + c * 7;
  float acc = 0.0f;
#pragma unroll
  for (int k = 0; k < 7; ++k) {
    int mm = m + k;
    acc += X[(long)no[mm >> shift] * F + (mm & (F - 1))];
  }
  out[idx] = acc * (1.0f / 7.0f);
}

// ---------------------------------------------------------------------------
// Upconv assemble + skip concat.
// y2 flat == Y flat (raw reshape of (N,7*oc) to (N*7,oc)).
// Rows < raw:  y2[ti[i]].  Rows >= raw: mean of adjacent columns (2c,2c+1)
// of row y2[di[2*ii + (2c>=oc)]]  (faithful to reshape(-1,oc,2).mean(-1)).
// Output is the concat buffer (upN x 2*oc): up part cols [0,oc), skip [oc,2oc).
// ---------------------------------------------------------------------------
__global__ void k_upcat(const float* __restrict__ Yflat, const int* __restrict__ ti,
                        const int* __restrict__ di, const float* __restrict__ skip,
                        float* __restrict__ out, int raw, int upN, int oc)
{
  int C2 = 2 * oc;
  int idx = blockIdx.x * blockDim.x + threadIdx.x;
  if (idx >= upN * C2) return;
  int i = idx / C2;
  int c = idx - i * C2;
  float v;
  if (c >= oc) {
    v = skip[(long)i * oc + (c - oc)];
  } else if (i < raw) {
    v = Yflat[(long)ti[i] * oc + c];
  } else {
    int ii = i - raw;
    int q  = (2 * c >= oc) ? 1 : 0;
    int j  = di[2 * ii + q];
    int b0 = 2 * c - q * oc;
    v = 0.5f * (Yflat[(long)j * oc + b0] + Yflat[(long)j * oc + b0 + 1]);
  }
  out[idx] = v;
}

// ---------------------------------------------------------------------------
// Final linear 32 -> 45 (45 not WMMA-friendly; scalar dot per output elem).
// ---------------------------------------------------------------------------
__global__ void k_final(const float* __restrict__ U, const float* __restrict__ oW,
                        const float* __restrict__ ob, float* __restrict__ out, int M)
{
  int idx = blockIdx.x * blockDim.x + threadIdx.x;
  if (idx >= M * 45) return;
  int i = idx / 45;
  int o = idx - i * 45;
  float acc = ob[o];
#pragma unroll
  for (int c = 0; c < 32; ++c) acc += U[(long)i * 32 + c] * oW[o * 32 + c];
  out[idx] = acc;
}

// ---------------------------------------------------------------------------
extern "C" void kernel_launch(void* const* d_in, const int* in_sizes, int n_in,
                              void* d_out, int out_size, void* d_ws, size_t ws_size,
                              hipStream_t stream)
{
  (void)in_sizes; (void)n_in; (void)out_size; (void)ws_size;

  const int Nlev[5] = {40962, 10242, 2562, 642, 162};
  const int ch[6]   = {45, 32, 64, 128, 256, 512};

  const float* x = (const float*)d_in[0];
  const int* no[5];
  for (int i = 0; i < 5; ++i) no[i] = (const int*)d_in[1 + i];
  const int* ut[4]; const int* ud[4];
  for (int j = 0; j < 4; ++j) { ut[j] = (const int*)d_in[6 + 2*j]; ud[j] = (const int*)d_in[7 + 2*j]; }

  int pi = 14;
  auto nf = [&]() { return (const float*)d_in[pi++]; };
  const float *dW1[6], *db1[6], *dg1[6], *de1[6], *dW2[6], *db2[6], *dg2[6], *de2[6];
  for (int i = 1; i <= 5; ++i) {
    dW1[i]=nf(); db1[i]=nf(); dg1[i]=nf(); de1[i]=nf();
    dW2[i]=nf(); db2[i]=nf(); dg2[i]=nf(); de2[i]=nf();
  }
  const float *uWu[5], *ubu[5], *uW1[5], *ub1[5], *ug1[5], *ue1[5], *uW2[5], *ub2[5], *ug2[5], *ue2[5];
  for (int j = 1; j <= 4; ++j) {
    uWu[j]=nf(); ubu[j]=nf(); uW1[j]=nf(); ub1[j]=nf(); ug1[j]=nf(); ue1[j]=nf();
    uW2[j]=nf(); ub2[j]=nf(); ug2[j]=nf(); ue2[j]=nf();
  }
  const float* oW = nf();
  const float* ob = nf();

  // ---- workspace bump allocator (floats) ----
  float* ws = (float*)d_ws;
  size_t off = 0;
  auto alloc = [&](size_t n) { float* p = ws + off; off += n; return p; };
  auto rnd32 = [](int k) { return (k + 31) & ~31; };

  auto maketr = [&](const float* W, int K, int N) -> float* {
    int Kp = rnd32(K);
    float* Wt = alloc((size_t)Kp * N);
    int tot = Kp * N;
    k_wt<<<(tot + 255) / 256, 256, 0, stream>>>(W, Wt, K, Kp, N);
    return Wt;
  };

  // transposed weights
  float *tW1[6], *tW2[6], *tWu[5], *tU1[5], *tU2[5];
  for (int i = 1; i <= 5; ++i) {
    tW1[i] = maketr(dW1[i], 7 * ch[i - 1], ch[i]);
    tW2[i] = maketr(dW2[i], 7 * ch[i],     ch[i]);
  }
  for (int j = 1; j <= 4; ++j) {
    int ic = ch[6 - j], oc = ch[5 - j];
    tWu[j] = maketr(uWu[j], ic, 7 * oc);
    tU1[j] = maketr(uW1[j], 7 * ic, oc);     // ic == 2*oc (concat width)
    tU2[j] = maketr(uW2[j], 7 * oc, oc);
  }

  // feature buffers
  float* x2 = alloc((size_t)40962 * 32);
  float* x3 = alloc((size_t)10242 * 64);
  float* x4 = alloc((size_t)2562 * 128);
  float* x5 = alloc((size_t)642 * 256);
  float* x6 = alloc((size_t)162 * 512);
  const size_t TSZ = (size_t)40962 * 64;
  float* Ta = alloc(TSZ);
  float* Tb = alloc(TSZ);
  float* Tc = alloc(TSZ);
  float* mean = alloc(512);
  float* var  = alloc(512);

  auto gemm = [&](const float* X, const int* noP, const float* Wt, const float* bias,
                  float* Y, int M, int K, int N, int ic, int ldY) {
    dim3 g((M + TILE_M - 1) / TILE_M, N / TILE_N);
    k_gemm<<<g, 256, 0, stream>>>(X, noP, Wt, bias, Y, M, K, rnd32(K), N, ic, ldY, 0);
  };
  auto bn = [&](float* Y, const float* gg, const float* ee, int M, int C) {
    k_bnstats<<<C, 256, 0, stream>>>(Y, mean, var, M, C);
    k_bnapply<<<(M * C + 255) / 256, 256, 0, stream>>>(Y, mean, var, gg, ee, M, C);
  };

  // ---- encoder ----
  // d1 (input x has ic=45)
  gemm(x,  no[0], tW1[1], db1[1], Ta, 40962, 7 * 45, 32, 45, 32);
  bn(Ta, dg1[1], de1[1], 40962, 32);
  gemm(Ta, no[0], tW2[1], db2[1], x2, 40962, 7 * 32, 32, 32, 32);
  bn(x2, dg2[1], de2[1], 40962, 32);

  float* sk[6] = {nullptr, x2, x3, x4, x5, x6};
  for (int i = 2; i <= 5; ++i) {
    int nOut = Nlev[i - 1];
    int cin  = ch[i - 1];
    int cout = ch[i];
    int sh = 0; while ((1 << sh) < cin) ++sh;
    k_pool<<<(nOut * cin + 255) / 256, 256, 0, stream>>>(sk[i - 1], no[i - 2], Ta, nOut, sh);
    gemm(Ta, no[i - 1], tW1[i], db1[i], Tb, nOut, 7 * cin, cout, cin, cout);
    bn(Tb, dg1[i], de1[i], nOut, cout);
    gemm(Tb, no[i - 1], tW2[i], db2[i], sk[i], nOut, 7 * cout, cout, cout, cout);
    bn(sk[i], dg2[i], de2[i], nOut, cout);
  }

  // ---- decoder ----
  float* bufYup[5] = {nullptr, Ta, Tc, Ta, Tc};
  float* bufT1 [5] = {nullptr, Tc, Ta, Tc, Tc};
  float* bufOut[5] = {nullptr, Ta, Tc, Ta, Ta};
  const float* skipArr[5] = {nullptr, x5, x4, x3, x2};
  const int rawArr[5] = {0, 162, 642, 2562, 10242};

  float* uin = x6;
  for (int j = 1; j <= 4; ++j) {
    int ic  = ch[6 - j], oc = ch[5 - j];
    int raw = rawArr[j];
    int upN = 4 * raw - 6;
    int Ncols = 7 * oc;
    // upconv GEMM: (raw x ic) @ Wu^T -> (raw x 7*oc)
    gemm(uin, nullptr, tWu[j], ubu[j], bufYup[j], raw, ic, Ncols, ic, Ncols);
    // assemble up rows + concat skip into (upN x 2*oc)
    k_upcat<<<((size_t)upN * 2 * oc + 255) / 256, 256, 0, stream>>>(
        bufYup[j], ut[j - 1], ud[j - 1], skipArr[j], Tb, raw, upN, oc);
    // double conv on concat (ic_cat = 2*oc)
    gemm(Tb, no[4 - j], tU1[j], ub1[j], bufT1[j], upN, 7 * 2 * oc, oc, 2 * oc, oc);
    bn(bufT1[j], ug1[j], ue1[j], upN, oc);
    gemm(bufT1[j], no[4 - j], tU2[j], ub2[j], bufOut[j], upN, 7 * oc, oc, oc, oc);
    bn(bufOut[j], ug2[j], ue2[j], upN, oc);
    uin = bufOut[j];
  }

  // ---- output linear 32 -> 45 ----
  k_final<<<(40962 * 45 + 255) / 256, 256, 0, stream>>>(uin, oW, ob, (float*)d_out, 40962);
}